// ExportModel_44702019617605
// MI455X (gfx1250) — compile-verified
//
#include <hip/hip_runtime.h>
#include <stdint.h>

#define NANCH 8400
#define NCLS  80
#define NSORT 16384
#define NTOP  1024
#define CONF_T 0.25f
#define IOU_T  0.45f
#define MAXDET 300
#define MAXWH  7680.0f
#define WS_STRIDE 8448   // padded per-array stride (floats) in workspace

typedef unsigned long long u64;
typedef unsigned int       u32;
typedef float v2f __attribute__((ext_vector_type(2)));
typedef float v8f __attribute__((ext_vector_type(8)));

// ---- LDS layout for nms_kernel (dynamic shared) ----
#define K_KEYS   0          // u64[16384] = 131072 B ; later reused as supp u32[1024*32]
#define K_TCONF  131072     // float[1024]
#define K_TCLS   135168     // float[1024]
#define K_RX0    139264     // float[1024] raw xyxy (gather dst)
#define K_RY0    143360
#define K_RX1    147456
#define K_RY1    151552
#define K_OX0    155648     // float[1024] class-offset xyxy
#define K_OY0    159744
#define K_OX1    163840
#define K_OY1    167936
#define K_AREA   172032     // float[1024]
#define K_KEEP   176128     // u32[32]
#define SMEM_TOTAL 176256

__device__ __forceinline__ u32 fkey(float f) {
  u32 u = __float_as_uint(f);
  return (u & 0x80000000u) ? ~u : (u | 0x80000000u);
}
__device__ __forceinline__ float funkey(u32 u) {
  return __uint_as_float((u & 0x80000000u) ? (u & 0x7FFFFFFFu) : ~u);
}
__device__ __forceinline__ u32 lds_off(const void* p) {
  // generic shared pointer: low 32 bits are the LDS byte offset (ISA flat->LDS rule)
  return (u32)(size_t)p;
}
// CDNA5 async global->LDS copy (ASYNCcnt tracked), portable via inline asm
__device__ __forceinline__ void async_gather_b32(u32 dst_lds, const float* src) {
  asm volatile("global_load_async_to_lds_b32 %0, %1, off"
               :: "v"(dst_lds), "v"((u64)(size_t)src)
               : "memory");
}
__device__ __forceinline__ void wait_async0() {
  asm volatile("s_wait_asynccnt 0x0" ::: "memory");
}

// ---------------- Kernel 1: per-anchor conf/argmax + xywh->xyxy ----------------
__global__ void prep_kernel(const float* __restrict__ preds,
                            float* __restrict__ conf, float* __restrict__ cls,
                            float* __restrict__ bx0, float* __restrict__ by0,
                            float* __restrict__ bx1, float* __restrict__ by1) {
  int a = blockIdx.x * 256 + threadIdx.x;
  if (a >= NANCH) return;
  float x = preds[a];
  float y = preds[NANCH + a];
  float w = preds[2 * NANCH + a];
  float h = preds[3 * NANCH + a];
  float best = preds[4 * NANCH + a];
  int bc = 0;
  for (int c2 = 1; c2 < NCLS; ++c2) {
    if (c2 + 8 < NCLS) __builtin_prefetch(&preds[(4 + c2 + 8) * NANCH + a], 0, 1);
    float s = preds[(4 + c2) * NANCH + a];
    if (s > best) { best = s; bc = c2; }   // first-max == jnp.argmax
  }
  conf[a] = best;
  cls[a]  = (float)bc;
  bx0[a] = x - 0.5f * w;
  by0[a] = y - 0.5f * h;
  bx1[a] = x + 0.5f * w;
  by1[a] = y + 0.5f * h;
}

// ---------------- Kernel 2: sort + gather + IoU(WMMA) + greedy NMS + emit ----------------
__global__ void __launch_bounds__(1024) nms_kernel(
    const float* __restrict__ wsConf, const float* __restrict__ wsCls,
    const float* __restrict__ wsBx0, const float* __restrict__ wsBy0,
    const float* __restrict__ wsBx1, const float* __restrict__ wsBy1,
    float* __restrict__ out) {
  extern __shared__ __align__(16) char smem[];
  u64* keys    = (u64*)(smem + K_KEYS);
  u32* suppM   = (u32*)(smem + K_KEYS);      // overlays keys after gather
  float* tconf = (float*)(smem + K_TCONF);
  float* tcls  = (float*)(smem + K_TCLS);
  float* rx0   = (float*)(smem + K_RX0);
  float* ry0   = (float*)(smem + K_RY0);
  float* rx1   = (float*)(smem + K_RX1);
  float* ry1   = (float*)(smem + K_RY1);
  float* ox0   = (float*)(smem + K_OX0);
  float* oy0   = (float*)(smem + K_OY0);
  float* ox1   = (float*)(smem + K_OX1);
  float* oy1   = (float*)(smem + K_OY1);
  float* areaL = (float*)(smem + K_AREA);
  u32* keepw   = (u32*)(smem + K_KEEP);

  const int tid  = threadIdx.x;
  const int lane = tid & 31;
  const int wv   = tid >> 5;

  // Phase 1: build sort keys (desc conf, asc index tie-break); pad with 0 (sorts last)
  for (int a = tid; a < NSORT; a += 1024) {
    if (a < NANCH) {
      float c = wsConf[a];
      float m = (c > CONF_T) ? c : -1.0f;            // masked_conf
      keys[a] = ((u64)fkey(m) << 32) | (u64)(0xFFFFFFFFu - (u32)a);
    } else {
      keys[a] = 0ull;
    }
  }

  // Phase 2: bitonic sort, descending (16384 elements, 128 KB LDS)
  for (u32 k = 2; k <= NSORT; k <<= 1) {
    for (u32 j = k >> 1; j > 0; j >>= 1) {
      __syncthreads();
      for (u32 t = tid; t < (NSORT / 2); t += 1024) {
        u32 i = ((t & ~(j - 1)) << 1) | (t & (j - 1));
        u32 l = i + j;
        u64 a = keys[i], b = keys[l];
        bool desc = ((i & k) == 0);
        if (desc ? (a < b) : (a > b)) { keys[i] = b; keys[l] = a; }
      }
    }
  }
  __syncthreads();

  // Phase 3: decode top-1024, async-gather cls + raw xyxy boxes into LDS
  {
    u64 key = keys[tid];
    float confv = funkey((u32)(key >> 32));
    u32 idx = 0xFFFFFFFFu - (u32)key;
    bool ok = (idx < NANCH);
    tconf[tid] = ok ? confv : -1.0f;
    u32 gidx = ok ? idx : 0u;
    async_gather_b32(lds_off(&tcls[tid]), wsCls + gidx);
    async_gather_b32(lds_off(&rx0[tid]),  wsBx0 + gidx);
    async_gather_b32(lds_off(&ry0[tid]),  wsBy0 + gidx);
    async_gather_b32(lds_off(&rx1[tid]),  wsBx1 + gidx);
    async_gather_b32(lds_off(&ry1[tid]),  wsBy1 + gidx);
    wait_async0();
  }
  __syncthreads();

  // Transform: class-offset boxes + areas; per-wave keep-init ballot
  {
    float clsv = tcls[tid];
    float x0 = rx0[tid], y0 = ry0[tid], x1 = rx1[tid], y1 = ry1[tid];
    float offv = clsv * MAXWH;
    ox0[tid] = x0 + offv; oy0[tid] = y0 + offv;
    ox1[tid] = x1 + offv; oy1[tid] = y1 + offv;
    areaL[tid] = (x1 - x0) * (y1 - y0);
    u32 kb = __builtin_amdgcn_ballot_w32(tconf[tid] > CONF_T);   // keep0
    if (lane == 0) keepw[wv] = kb;
  }
  __syncthreads();

  // Phase 4: suppression bit-matrix, 16x16 tiles per wave.
  // area_i + area_j per tile via V_WMMA_F32_16X16X4_F32 outer-product (A=ones col K=0,
  // B=area_j row K=0, C=area_i broadcast); intersection via VALU; pack via wave32 ballot.
  {
    const int hi8 = (lane >= 16) ? 8 : 0;
    const int lj  = lane & 15;
    for (int kk = 0; kk < 128; ++kk) {
      int t  = wv * 128 + kk;        // 4096 tiles total
      int ti = t >> 6, tj = t & 63;
      int j  = tj * 16 + lj;
      float jx0 = ox0[j], jy0 = oy0[j], jx1 = ox1[j], jy1 = oy1[j];
      v2f av, bv;
      av.x = (lane < 16) ? 1.0f : 0.0f;        av.y = 0.0f;   // A[i][0] = 1
      bv.x = (lane < 16) ? areaL[j] : 0.0f;    bv.y = 0.0f;   // B[0][j] = area_j
      v8f cv;
#pragma unroll
      for (int v = 0; v < 8; ++v) cv[v] = areaL[ti * 16 + hi8 + v];   // C = area_i
      v8f dv = __builtin_amdgcn_wmma_f32_16x16x4_f32(
          false, av, false, bv, (short)0, cv, false, false);
#pragma unroll
      for (int v = 0; v < 8; ++v) {
        int i = ti * 16 + hi8 + v;
        float ltx = fmaxf(ox0[i], jx0);
        float lty = fmaxf(oy0[i], jy0);
        float rbx = fminf(ox1[i], jx1);
        float rby = fminf(oy1[i], jy1);
        float inter = fmaxf(rbx - ltx, 0.0f) * fmaxf(rby - lty, 0.0f);
        float iou = inter / (dv[v] - inter + 1e-7f);   // dv = area_i + area_j
        bool pred = (iou > IOU_T) && (j > i);
        u32 m = __builtin_amdgcn_ballot_w32(pred);     // lo16: row ti*16+v, hi16: +8
        if (lane == 0) {
          unsigned short* sp = (unsigned short*)suppM;
          sp[(ti * 16 + v) * 64 + tj]     = (unsigned short)(m & 0xFFFFu);
          sp[(ti * 16 + v + 8) * 64 + tj] = (unsigned short)(m >> 16);
        }
      }
    }
  }
  __syncthreads();

  // Phase 5: greedy sequential scan on wave 0 (keep mask: one u32 per lane)
  if (tid < 32) {
    u32 kw = keepw[tid];
    for (int i = 0; i < NTOP; ++i) {
      u32 ww = __builtin_amdgcn_readlane(kw, i >> 5);
      if ((ww >> (i & 31)) & 1u) kw &= ~suppM[i * 32 + tid];
    }
    keepw[tid] = kw;
  }
  __syncthreads();

  // Phase 6: emit top-300 kept (already in descending-conf order)
  for (int t = tid; t < MAXDET * 6; t += 1024) out[t] = 0.0f;
  __syncthreads();
  {
    int i = tid;
    u32 w = keepw[i >> 5];
    if ((w >> (i & 31)) & 1u) {
      int rank = __popc(w & ((1u << (i & 31)) - 1u));
      for (int m = 0; m < (i >> 5); ++m) rank += __popc(keepw[m]);
      if (rank < MAXDET) {
        float* r = out + rank * 6;
        r[0] = rx0[i]; r[1] = ry0[i]; r[2] = rx1[i]; r[3] = ry1[i];
        r[4] = tconf[i]; r[5] = tcls[i];
      }
    }
  }
}

extern "C" void kernel_launch(void* const* d_in, const int* in_sizes, int n_in,
                              void* d_out, int out_size, void* d_ws, size_t ws_size,
                              hipStream_t stream) {
  const float* preds = (const float*)d_in[0];   // (32,84,8400); only batch 0 needed
  float* wsF  = (float*)d_ws;
  float* conf = wsF + 0 * WS_STRIDE;
  float* cls  = wsF + 1 * WS_STRIDE;
  float* bx0  = wsF + 2 * WS_STRIDE;
  float* by0  = wsF + 3 * WS_STRIDE;
  float* bx1  = wsF + 4 * WS_STRIDE;
  float* by1  = wsF + 5 * WS_STRIDE;

  prep_kernel<<<(NANCH + 255) / 256, 256, 0, stream>>>(preds, conf, cls, bx0, by0, bx1, by1);

  (void)hipFuncSetAttribute((const void*)nms_kernel,
                            hipFuncAttributeMaxDynamicSharedMemorySize, SMEM_TOTAL);
  nms_kernel<<<1, 1024, SMEM_TOTAL, stream>>>(conf, cls, bx0, by0, bx1, by1, (float*)d_out);
}